// GAT_23794118820071
// MI455X (gfx1250) — compile-verified
//
#include <hip/hip_runtime.h>
#include <math.h>

// ---------------------------------------------------------------------------
// GAT (2-layer, 4 heads, 128 features) for gfx1250.
// GEMM uses V_WMMA_F32_16X16X4_F32 (full fp32 matrix pipe, matches reference).
// ---------------------------------------------------------------------------

typedef float v2f __attribute__((ext_vector_type(2)));
typedef float v8f __attribute__((ext_vector_type(8)));

#define F 128          // feature width (in = out = 128)
#define HEADS 4
#define CPH 32         // channels per head
#define NEG_SLOPE 0.2f

// -------------------- helpers --------------------

__device__ inline void atomicMaxFloat(float* addr, float val) {
  // sign-aware monotonic mapping: works with addr initialized to -inf bits
  if (val >= 0.0f) {
    atomicMax((int*)addr, __float_as_int(val));
  } else {
    atomicMin((unsigned int*)addr, __float_as_uint(val));
  }
}

// -------------------- 1) GEMM: H = X @ W  (row-major, 128x128 W) -----------
// one wave -> one 16-row tile, loops over all 8 column tiles.
// W staged in LDS (64 KB) once per block.
__global__ __launch_bounds__(256) void k_gemm_wmma(const float* __restrict__ X,
                                                   const float* __restrict__ W,
                                                   float* __restrict__ H,
                                                   int n) {
  __shared__ float Wl[F * F];   // 64 KB
  // cooperative load of W into LDS, 128-bit chunks
  for (int i = threadIdx.x; i < (F * F) / 4; i += 256) {
    ((float4*)Wl)[i] = ((const float4*)W)[i];
  }
  __syncthreads();

  const int wave = threadIdx.x >> 5;
  const int lane = threadIdx.x & 31;
  const int tile = blockIdx.x * 8 + wave;
  const int m0   = tile * 16;
  if (m0 >= n) return;                    // wave-uniform: EXEC stays all-ones

  const int half = lane >> 4;             // 0: K even pair, 1: K odd pair
  const int l16  = lane & 15;
  const int row  = m0 + l16;
  const bool rowOK = (row < n);

  v8f acc[8];
#pragma unroll
  for (int t = 0; t < 8; ++t)
#pragma unroll
    for (int j = 0; j < 8; ++j) acc[t][j] = 0.0f;

  for (int kk = 0; kk < F / 4; ++kk) {    // 32 K-steps of 4
    const int k0 = kk * 4 + half * 2;
    v2f a;
    if (rowOK) {
      a = *(const v2f*)(X + (size_t)row * F + k0);   // 8B aligned (k0 even)
    } else {
      a.x = 0.0f; a.y = 0.0f;
    }
#pragma unroll
    for (int nt = 0; nt < 8; ++nt) {
      const int col = nt * 16 + l16;
      v2f b;
      b.x = Wl[k0 * F + col];
      b.y = Wl[(k0 + 1) * F + col];
      acc[nt] = __builtin_amdgcn_wmma_f32_16x16x4_f32(
          /*neg_a=*/false, a, /*neg_b=*/false, b,
          /*c_mod=*/(short)0, acc[nt], /*reuse_a=*/false, /*reuse_b=*/false);
    }
  }

  // C/D layout: VGPR r, lanes 0-15 -> M=r, lanes 16-31 -> M=8+r; N = l16+nt*16
#pragma unroll
  for (int nt = 0; nt < 8; ++nt) {
#pragma unroll
    for (int r = 0; r < 8; ++r) {
      const int orow = m0 + r + half * 8;
      if (orow < n) H[(size_t)orow * F + nt * 16 + l16] = acc[nt][r];
    }
  }
}

// -------------------- 2) per-node attention coefficients -------------------
// one thread per (node, head): 32-wide dots with a_src / a_dst
__global__ void k_coef(const float* __restrict__ H,
                       const float* __restrict__ a_src,
                       const float* __restrict__ a_dst,
                       float* __restrict__ ssrc, float* __restrict__ sdst,
                       int n) {
  int tid = blockIdx.x * blockDim.x + threadIdx.x;
  if (tid >= n * HEADS) return;
  const int node = tid >> 2;
  const int h    = tid & 3;
  const float* hp = H + (size_t)node * F + h * CPH;
  float ss = 0.0f, sd = 0.0f;
#pragma unroll
  for (int i = 0; i < CPH; ++i) {
    const float v = hp[i];
    ss += v * a_src[h * CPH + i];
    sd += v * a_dst[h * CPH + i];
  }
  ssrc[tid] = ss;
  sdst[tid] = sd;
}

// -------------------- 3) init accumulators ---------------------------------
__global__ void k_init(float* __restrict__ agg, float* __restrict__ amax,
                       float* __restrict__ denom, int n) {
  int tid = blockIdx.x * blockDim.x + threadIdx.x;
  if (tid < n * F) agg[tid] = 0.0f;
  if (tid < n * HEADS) {
    amax[tid]  = __uint_as_float(0xFF800000u);   // -inf
    denom[tid] = 0.0f;
  }
}

// -------------------- 4) alpha = leaky_relu(s_src[s]+s_dst[d]); segmax -----
__global__ void k_edge_alpha(const int* __restrict__ ei, int E, int n,
                             const float* __restrict__ ssrc,
                             const float* __restrict__ sdst,
                             float* __restrict__ alpha,
                             float* __restrict__ amax) {
  int tid = blockIdx.x * blockDim.x + threadIdx.x;
  const int etot = E + n;
  if (tid >= etot * HEADS) return;
  const int e = tid >> 2;
  const int h = tid & 3;
  int s, d;
  if (e < E) { s = ei[e]; d = ei[E + e]; } else { s = d = e - E; }
  float al = ssrc[s * HEADS + h] + sdst[d * HEADS + h];
  al = (al > 0.0f) ? al : NEG_SLOPE * al;
  alpha[tid] = al;
  atomicMaxFloat(&amax[d * HEADS + h], al);
}

// -------------------- 5) ex = exp(alpha - amax[d]); segsum -----------------
__global__ void k_edge_exp(const int* __restrict__ ei, int E, int n,
                           float* __restrict__ alpha,
                           const float* __restrict__ amax,
                           float* __restrict__ denom) {
  int tid = blockIdx.x * blockDim.x + threadIdx.x;
  const int etot = E + n;
  if (tid >= etot * HEADS) return;
  const int e = tid >> 2;
  const int h = tid & 3;
  int d;
  if (e < E) { d = ei[E + e]; } else { d = e - E; }
  const float ex = expf(alpha[tid] - amax[d * HEADS + h]);
  alpha[tid] = ex;
  atomicAdd(&denom[d * HEADS + h], ex);
}

// -------------------- 6) weighted scatter-add ------------------------------
// one wave per edge; lane = channel within head, loop over 4 heads.
__global__ __launch_bounds__(256) void k_edge_scatter(
    const int* __restrict__ ei, int E, int n,
    const float* __restrict__ H, const float* __restrict__ alpha,
    const float* __restrict__ denom, float* __restrict__ agg) {
  const int wid  = (blockIdx.x * blockDim.x + threadIdx.x) >> 5;
  const int lane = threadIdx.x & 31;
  const int etot = E + n;
  if (wid >= etot) return;
  int s, d;
  if (wid < E) { s = ei[wid]; d = ei[E + wid]; } else { s = d = wid - E; }
#pragma unroll
  for (int h = 0; h < HEADS; ++h) {
    const float attn = alpha[(size_t)wid * HEADS + h] /
                       (denom[d * HEADS + h] + 1e-16f);
    const float val = H[(size_t)s * F + h * CPH + lane] * attn;
    atomicAdd(&agg[(size_t)d * F + h * CPH + lane], val);
  }
}

// -------------------- 7) epilogue: y = relu(agg + b) -----------------------
__global__ void k_epilogue(const float* __restrict__ agg,
                           const float* __restrict__ bias,
                           float* __restrict__ y, int n) {
  int tid = blockIdx.x * blockDim.x + threadIdx.x;
  if (tid >= n * F) return;
  const float v = agg[tid] + bias[tid & (F - 1)];
  y[tid] = (v > 0.0f) ? v : 0.0f;
}

// ---------------------------------------------------------------------------

static inline int cdiv(int a, int b) { return (a + b - 1) / b; }

static void run_layer(const float* X, const int* ei, int E, int n,
                      const float* W, const float* a_src, const float* a_dst,
                      const float* bias,
                      float* H, float* ALPH, float* SSRC, float* SDST,
                      float* AMAX, float* DEN, float* AGG, float* Y,
                      hipStream_t stream) {
  const int etot = E + n;
  const int tiles = cdiv(n, 16);
  k_gemm_wmma<<<cdiv(tiles, 8), 256, 0, stream>>>(X, W, H, n);
  k_coef<<<cdiv(n * HEADS, 256), 256, 0, stream>>>(H, a_src, a_dst, SSRC, SDST, n);
  k_init<<<cdiv(n * F, 256), 256, 0, stream>>>(AGG, AMAX, DEN, n);
  k_edge_alpha<<<cdiv(etot * HEADS, 256), 256, 0, stream>>>(ei, E, n, SSRC, SDST,
                                                            ALPH, AMAX);
  k_edge_exp<<<cdiv(etot * HEADS, 256), 256, 0, stream>>>(ei, E, n, ALPH, AMAX, DEN);
  k_edge_scatter<<<cdiv(etot, 8), 256, 0, stream>>>(ei, E, n, H, ALPH, DEN, AGG);
  k_epilogue<<<cdiv(n * F, 256), 256, 0, stream>>>(AGG, bias, Y, n);
}

extern "C" void kernel_launch(void* const* d_in, const int* in_sizes, int n_in,
                              void* d_out, int out_size, void* d_ws, size_t ws_size,
                              hipStream_t stream) {
  const float* x      = (const float*)d_in[0];
  const int*   ei     = (const int*)d_in[1];    // [2, E] flattened
  const float* W1     = (const float*)d_in[2];
  const float* a_src1 = (const float*)d_in[3];
  const float* a_dst1 = (const float*)d_in[4];
  const float* b1     = (const float*)d_in[5];
  const float* W2     = (const float*)d_in[6];
  const float* a_src2 = (const float*)d_in[7];
  const float* a_dst2 = (const float*)d_in[8];
  const float* b2     = (const float*)d_in[9];
  float* out = (float*)d_out;

  const int n = in_sizes[0] / F;     // 50000
  const int E = in_sizes[1] / 2;     // 800000
  const int etot = E + n;

  // workspace carving (floats)
  float* ws   = (float*)d_ws;
  float* H    = ws;                             // n*F
  float* ALPH = H + (size_t)n * F;              // etot*HEADS
  float* SSRC = ALPH + (size_t)etot * HEADS;    // n*HEADS
  float* SDST = SSRC + (size_t)n * HEADS;       // n*HEADS
  float* AMAX = SDST + (size_t)n * HEADS;       // n*HEADS
  float* DEN  = AMAX + (size_t)n * HEADS;       // n*HEADS
  float* Y1   = DEN + (size_t)n * HEADS;        // n*F (layer-1 output)
  float* AGG1 = Y1 + (size_t)n * F;             // n*F

  // layer 1: x -> Y1
  run_layer(x, ei, E, n, W1, a_src1, a_dst1, b1,
            H, ALPH, SSRC, SDST, AMAX, DEN, AGG1, Y1, stream);
  // layer 2: Y1 -> out (aggregate directly into d_out, relu in place)
  run_layer(Y1, ei, E, n, W2, a_src2, a_dst2, b2,
            H, ALPH, SSRC, SDST, AMAX, DEN, out, out, stream);
}